// MetaQDA_FB_4097398800825
// MI455X (gfx1250) — compile-verified
//
#include <hip/hip_runtime.h>
#include <hip/hip_bf16.h>

#define D_   640
#define C_   64
#define S_   16
#define Q_   2048
#define REG_ 0.3f
#define EPS_ 1e-6f

typedef __attribute__((ext_vector_type(2))) float    v2f;
typedef __attribute__((ext_vector_type(8))) float    v8f;
typedef __attribute__((ext_vector_type(4))) unsigned u32x4;
typedef __attribute__((ext_vector_type(8))) int      i32x8;
typedef __attribute__((ext_vector_type(4))) int      i32x4;

__device__ __forceinline__ v8f wmma_f32_4(v2f a, v2f b, v8f c) {
    // (neg_a, A, neg_b, B, c_mod, C, reuse_a, reuse_b)
    return __builtin_amdgcn_wmma_f32_16x16x4_f32(false, a, false, b, (short)0, c,
                                                 false, false);
}

// Truncate a generic pointer to an LDS byte offset (low 32 bits of the
// shared-aperture generic address are the DS-style offset).
__device__ __forceinline__ unsigned lds_off_of(const void* p) {
    return (unsigned)(unsigned long long)p;
}

// ---------------------------------------------------------------- kernel 1
__global__ void k_build_L(const float* __restrict__ tdiag,
                          const float* __restrict__ tlow,
                          float* __restrict__ L) {
    int idx = blockIdx.x * blockDim.x + threadIdx.x;
    if (idx >= D_ * D_) return;
    int i = idx / D_, j = idx - i * D_;
    float v = 0.f;
    if (i == j)      v = fabsf(tdiag[i]);
    else if (j < i)  v = tlow[idx];
    L[idx] = v;
}

// ---------------------------------------------------------------- kernel 2
__global__ void k_build_mu(const float* __restrict__ Xs,
                           const float* __restrict__ m,
                           const float* __restrict__ kappa_p,
                           float* __restrict__ mu) {
    int idx = blockIdx.x * blockDim.x + threadIdx.x;
    if (idx >= C_ * D_) return;
    int c = idx / D_, j = idx - c * D_;
    float kap = fabsf(kappa_p[0]) + EPS_;
    float s = 0.f;
    const float* Xc = Xs + (size_t)c * S_ * D_;
    for (int t = 0; t < S_; ++t) s += Xc[t * D_ + j];
    float mean = s * (1.f / (float)S_);
    float Nj = (float)S_;
    mu[idx] = (kap / (kap + Nj)) * m[j] + (Nj / (kap + Nj)) * mean;
}

// ---------------------------------------------------------------- kernel 3
// LLT = L @ L^T, one wave per 16x16 output tile, f32 WMMA K=4 steps.
__global__ __launch_bounds__(32) void k_llt_wmma(const float* __restrict__ L,
                                                 float* __restrict__ LLT) {
    int tm = blockIdx.x, tn = blockIdx.y;
    int lane = threadIdx.x;
    int hi = lane >> 4, lo = lane & 15;
    int mrow = tm * 16 + lo;   // A-operand row (M) held by this lane
    int ncol = tn * 16 + lo;   // B-operand col (N) held by this lane
    v8f acc = {};
    for (int k = 0; k < D_; k += 4) {
        int k0 = k + 2 * hi;
        v2f a, b;
        a.x = L[mrow * D_ + k0];
        a.y = L[mrow * D_ + k0 + 1];
        // B[kk][n] = (L^T)[kk][n] = L[n][kk]
        b.x = L[ncol * D_ + k0];
        b.y = L[ncol * D_ + k0 + 1];
        acc = wmma_f32_4(a, b, acc);
    }
    for (int r = 0; r < 8; ++r)
        LLT[(tm * 16 + r + 8 * hi) * D_ + tn * 16 + lo] = acc[r];
}

// ---------------------------------------------------------------- kernel 4
// sigma[c] = scale * (LLT + Xg_c^T Xg_c + kap*m m^T - (kap+N) mu_c mu_c^T)
__global__ __launch_bounds__(32) void k_sigma_wmma(
        const float* __restrict__ Xs, const float* __restrict__ LLT,
        const float* __restrict__ m,  const float* __restrict__ mu,
        const float* __restrict__ kappa_p, const float* __restrict__ nu_p,
        float* __restrict__ sigma) {
    int tm = blockIdx.x, tn = blockIdx.y, c = blockIdx.z;
    int lane = threadIdx.x;
    int hi = lane >> 4, lo = lane & 15;
    float kap = fabsf(kappa_p[0]) + EPS_;
    float nu  = fmaxf(nu_p[0], (float)(D_ - 1) + EPS_);
    float Nj  = (float)S_;
    float scale = (kap + Nj + 1.f) / ((nu + Nj - (float)D_ + 1.f) * (kap + Nj));

    const float* Xc = Xs + (size_t)c * S_ * D_;
    int mrow = tm * 16 + lo;
    int ncol = tn * 16 + lo;
    v8f acc = {};
    for (int k = 0; k < S_; k += 4) {           // K = shots (16) -> 4 steps
        int k0 = k + 2 * hi;
        v2f a, b;
        a.x = Xc[(k0)     * D_ + mrow];
        a.y = Xc[(k0 + 1) * D_ + mrow];
        b.x = Xc[(k0)     * D_ + ncol];
        b.y = Xc[(k0 + 1) * D_ + ncol];
        acc = wmma_f32_4(a, b, acc);
    }
    const float* muc = mu + (size_t)c * D_;
    float* Sc = sigma + (size_t)c * D_ * D_;
    for (int r = 0; r < 8; ++r) {
        int i = tm * 16 + r + 8 * hi;
        int j = tn * 16 + lo;
        float v = acc[r] + LLT[i * D_ + j] + kap * m[i] * m[j]
                - (kap + Nj) * muc[i] * muc[j];
        Sc[i * D_ + j] = v * scale;
    }
}

// ---------------------------------------------------------------- kernel 5
// In-place Gauss-Jordan inverse of SPD matrix per class + logdet from pivots.
__global__ __launch_bounds__(1024) void k_gj_invert(float* __restrict__ sigma,
                                                    float* __restrict__ logdet) {
    __shared__ float colk[D_];
    __shared__ float rowk[D_];
    int c = blockIdx.x;
    float* W = sigma + (size_t)c * D_ * D_;
    int tid = threadIdx.x;
    float ld = 0.f;
    for (int k = 0; k < D_; ++k) {
        // snapshot old column k
        for (int i = tid; i < D_; i += 1024) colk[i] = W[i * D_ + k];
        __syncthreads();
        float p = colk[k];
        float invp = 1.f / p;
        // new row k: W[k][k]=1/p, W[k][j]=W[k][j]/p
        for (int j = tid; j < D_; j += 1024) {
            float v = (j == k) ? invp : W[k * D_ + j] * invp;
            rowk[j] = v;
            W[k * D_ + j] = v;
        }
        __syncthreads();
        // all other rows: W[i][k] = -f/p, W[i][j] -= f * rowk[j]
        for (int idx = tid; idx < D_ * D_; idx += 1024) {
            int i = idx / D_;
            if (i == k) continue;
            int j = idx - i * D_;
            float f = colk[i];
            W[idx] = (j == k) ? (-f * invp) : (W[idx] - f * rowk[j]);
        }
        if (tid == 0) ld += logf(p);
        __syncthreads();
    }
    if (tid == 0) logdet[c] = ld;
}

// ---------------------------------------------------------------- kernel 6
__global__ void k_blend_reg(float* __restrict__ sigma) {
    size_t idx = (size_t)blockIdx.x * blockDim.x + threadIdx.x;
    if (idx >= (size_t)C_ * D_ * D_) return;
    size_t rem = idx % ((size_t)D_ * D_);
    int i = (int)(rem / D_), j = (int)(rem - (size_t)i * D_);
    float v = sigma[idx] * (1.f - REG_) + ((i == j) ? REG_ : 0.f);
    sigma[idx] = v;
}

// ---------------------------------------------------------------- kernel 7
__global__ void k_bias(const float* __restrict__ logdet,
                       const float* __restrict__ nu_p,
                       float* __restrict__ bias) {
    int c = threadIdx.x;
    if (c >= C_) return;
    float nu = fmaxf(nu_p[0], (float)(D_ - 1) + EPS_);
    float common = nu + (float)S_ + 1.f - (float)D_;
    bias[c] = lgammaf(0.5f * (common + (float)D_)) - lgammaf(0.5f * common)
            - 0.5f * (float)D_ * logf(common) - 0.5f * logdet[c];
}

// ---------------------------------------------------------------- kernel 8
// Fused maha + logits. Grid (C, Q/32). 32 query rows staged in LDS as diff;
// the 640x16 A^-1 column slab for each j-tile is brought in by the Tensor
// Data Mover (tensor_load_to_lds), double-buffered and synchronized with
// s_wait_tensorcnt. Two waves each own 16 query rows; f32 WMMA over the
// slab; shfl reduction; log1p epilogue.
#define QT 32
#define JT 16
__global__ __launch_bounds__(64) void k_maha_logits(
        const float* __restrict__ Xq, const float* __restrict__ mu,
        const float* __restrict__ Ainv, const float* __restrict__ bias,
        const float* __restrict__ nu_p, float* __restrict__ logits) {
    __shared__ float diff[QT * D_];                       // 80 KB
    __shared__ float bsl[2][D_ * JT];                     // 2 x 40 KB TDM slabs
    int c  = blockIdx.x;
    int qb = blockIdx.y * QT;
    int tid = threadIdx.x;                                // 0..63
    const float* muc = mu + (size_t)c * D_;
    for (int idx = tid; idx < QT * D_; idx += 64) {
        int r = idx / D_, j = idx - r * D_;
        diff[idx] = Xq[(size_t)(qb + r) * D_ + j] - muc[j];
    }

    const float* A = Ainv + (size_t)c * D_ * D_;
    const unsigned lds0 = lds_off_of(&bsl[0][0]);
    const unsigned lds1 = lds_off_of(&bsl[1][0]);
    const bool issuer = (tid < 32);                       // wave 0 drives the TDM

    // TDM descriptor: 2D tile, rows k=0..639, 16 contiguous f32 columns at jt.
    auto tdm_issue = [&](int jt, unsigned lds_addr) {
        unsigned long long ga = (unsigned long long)(const void*)(A + jt);
        u32x4 g0 = { 1u,                                  // count=1 (valid D#)
                     lds_addr,                            // lds_addr (bytes)
                     (unsigned)ga,                        // global_addr[31:0]
                     (unsigned)(ga >> 32) | (2u << 30) }; // addr[56:32] | type=2
        i32x8 g1 = { 0x00020000,                          // data_size=4B
                     (int)(D_ << 16),                     // tensor_dim0 = 640
                     (int)(D_ << 16),                     // tensor_dim1 = 640
                     (int)(JT << 16),                     // tile_dim0   = 16
                     D_,                                  // tile_dim1   = 640
                     D_,                                  // dim0_stride = 640
                     0, 0 };
        i32x4 gz4 = { 0, 0, 0, 0 };                       // 2D: groups 2/3 unused
        i32x8 gz8 = { 0, 0, 0, 0, 0, 0, 0, 0 };
        __builtin_amdgcn_tensor_load_to_lds(g0, g1, gz4, gz4, gz8, 0);
    };

    if (issuer) tdm_issue(0, lds0);                       // prefetch first slab

    int wave = tid >> 5;
    int lane = tid & 31;
    int hi = lane >> 4, lo = lane & 15;
    const float* dW = diff + wave * 16 * D_;              // this wave's 16 rows

    float mah[8];
    for (int r = 0; r < 8; ++r) mah[r] = 0.f;

    int buf = 0;
    for (int jt = 0; jt < D_; jt += JT, buf ^= 1) {
        if (issuer) {
            if (jt + JT < D_) {
                tdm_issue(jt + JT, buf ? lds0 : lds1);    // prefetch next slab
                __builtin_amdgcn_s_wait_tensorcnt(1);     // current slab landed
            } else {
                __builtin_amdgcn_s_wait_tensorcnt(0);
            }
        }
        __syncthreads();                                  // slab visible to both waves
        const float* B = &bsl[buf][0];                    // B[k][n] = A[k][jt+n]
        v8f acc = {};
        for (int k = 0; k < D_; k += 4) {
            int k0 = k + 2 * hi;
            v2f a, b;
            a.x = dW[lo * D_ + k0];                       // diff[M=lo][K]
            a.y = dW[lo * D_ + k0 + 1];
            b.x = B[(k0)     * JT + lo];                  // A[K][N=jt+lo] from LDS
            b.y = B[(k0 + 1) * JT + lo];
            acc = wmma_f32_4(a, b, acc);
        }
        for (int r = 0; r < 8; ++r)                       // P ∘ diff, row-sum
            mah[r] += acc[r] * dW[(r + 8 * hi) * D_ + jt + lo];
        __syncthreads();                                  // done with slab[buf]
    }

    float nu = fmaxf(nu_p[0], (float)(D_ - 1) + EPS_);
    float common = nu + (float)S_ + 1.f - (float)D_;
    float bc = bias[c];
    for (int r = 0; r < 8; ++r) {
        float v = mah[r];
        // reduce over the 16 lanes (same row group) holding N=0..15
        for (int mask = 1; mask <= 8; mask <<= 1)
            v += __shfl_xor(v, mask, 32);
        if (lo == 0) {
            int q = qb + wave * 16 + r + 8 * hi;
            logits[(size_t)q * C_ + c] =
                bc - 0.5f * (common + (float)D_) * log1pf(v / common);
        }
    }
}

// ---------------------------------------------------------------- launch
extern "C" void kernel_launch(void* const* d_in, const int* in_sizes, int n_in,
                              void* d_out, int out_size, void* d_ws, size_t ws_size,
                              hipStream_t stream) {
    (void)in_sizes; (void)n_in; (void)out_size; (void)ws_size;
    const float* Xs    = (const float*)d_in[0];
    /* d_in[1] = y (int64) — support is already class-sorted by construction */
    const float* Xq    = (const float*)d_in[2];
    const float* m     = (const float*)d_in[3];
    const float* kappa = (const float*)d_in[4];
    const float* nu    = (const float*)d_in[5];
    const float* tdiag = (const float*)d_in[6];
    const float* tlow  = (const float*)d_in[7];
    float* out = (float*)d_out;

    float* ws     = (float*)d_ws;
    float* L      = ws;                       // 409600
    float* LLT    = ws + 409600;              // 409600
    float* mu     = ws + 819200;              // 40960
    float* sigma  = ws + 860160;              // C*D*D = 26214400
    float* logdet = sigma + (size_t)C_ * D_ * D_;   // 64
    float* bias   = logdet + C_;                    // 64

    k_build_L<<<(D_ * D_ + 255) / 256, 256, 0, stream>>>(tdiag, tlow, L);
    k_build_mu<<<(C_ * D_ + 255) / 256, 256, 0, stream>>>(Xs, m, kappa, mu);
    k_llt_wmma<<<dim3(D_ / 16, D_ / 16), 32, 0, stream>>>(L, LLT);
    k_sigma_wmma<<<dim3(D_ / 16, D_ / 16, C_), 32, 0, stream>>>(
        Xs, LLT, m, mu, kappa, nu, sigma);
    k_gj_invert<<<C_, 1024, 0, stream>>>(sigma, logdet);
    {
        size_t tot = (size_t)C_ * D_ * D_;
        k_blend_reg<<<(unsigned)((tot + 255) / 256), 256, 0, stream>>>(sigma);
    }
    k_bias<<<1, C_, 0, stream>>>(logdet, nu, bias);
    k_maha_logits<<<dim3(C_, Q_ / QT), 64, 0, stream>>>(
        Xq, mu, sigma, bias, nu, out);
}